// NCDE_Encoder_52037823758376
// MI455X (gfx1250) — compile-verified
//
#include <hip/hip_runtime.h>
#include <hip/hip_bf16.h>
#include <math.h>

// Problem constants (from reference): B=2048, L=128, C=32, H=64, O=16, N_SUB=2
#define B_SZ 2048
#define L_SZ 128
#define C_SZ 32
#define H_SZ 64
#define O_SZ 16
#define NI   (L_SZ - 1)   // 127 intervals

typedef __attribute__((ext_vector_type(2))) float v2f;
typedef __attribute__((ext_vector_type(8))) float v8f;

__device__ __forceinline__ float fast_tanh(float x) {
#if defined(__AMDGCN__) && __has_builtin(__builtin_amdgcn_tanhf)
    return __builtin_amdgcn_tanhf(x);       // v_tanh_f32 (TRANS op on gfx1250)
#else
    float e = __expf(-2.0f * fabsf(x));     // v_exp_f32 path fallback
    float r = (1.0f - e) / (1.0f + e);
    return copysignf(r, x);
#endif
}

// ---------------------------------------------------------------------------
// Kernel 1: z0 init + sequential RK4 scan. One wave32 per batch element.
// Lane `l` owns state components h0=l and h1=l+32; lane `l` also owns channel
// c=l for the dXdt contraction (C==32==wave width).
//
// LDS layout is TRANSPOSED to [c][h] so the inner loop (fixed c, h=lane) hits
// consecutive dwords -> conflict-free across all 64 banks. W2's two rows are
// packed as float2 -> one ds_load_b64 + one ds_load_b32 per tanh argument.
//
// Writes pre-tanh z history: z_hist[b][l][h], l = 0..127.
// ---------------------------------------------------------------------------
__global__ __launch_bounds__(256) void ncde_scan_kernel(
    const float* __restrict__ coeffs,   // (B, 127, 128) packed [a,b,2c,3d]
    const float* __restrict__ t,        // (128,)
    const float* __restrict__ W_init,   // (32, 64)
    const float* __restrict__ b_init,   // (64,)
    const float* __restrict__ W1,       // (64, 2)
    const float* __restrict__ b1,       // (2,)
    const float* __restrict__ W2,       // (2, 2048) row-major [j][h*32+c]
    const float* __restrict__ b2,       // (2048,)   [h*32+c]
    float* __restrict__ z_hist)         // (B, 128, 64)
{
    __shared__ float  sWinit[C_SZ * H_SZ];      // 2048 (8 KB), [c][h]
    __shared__ float  sW1[H_SZ * 2];            // 128
    __shared__ float  sb1[2];
    __shared__ float2 sPack[C_SZ * H_SZ];       // (W2[0],W2[1]) transposed [c][h] (16 KB)
    __shared__ float  sB2t[C_SZ * H_SZ];        // b2 transposed [c][h] (8 KB)
    __shared__ float  sT[L_SZ];

    const int tid = threadIdx.x;
    for (int i = tid; i < C_SZ * H_SZ; i += blockDim.x) sWinit[i] = W_init[i];
    for (int i = tid; i < H_SZ * 2;    i += blockDim.x) sW1[i]    = W1[i];
    for (int i = tid; i < C_SZ * H_SZ; i += blockDim.x) {
        const int c = i >> 6;          // i = c*64 + h
        const int h = i & 63;
        const int src = h * C_SZ + c;  // original [h][c] index
        sPack[i] = make_float2(W2[src], W2[2048 + src]);
        sB2t[i]  = b2[src];
    }
    for (int i = tid; i < L_SZ; i += blockDim.x) sT[i] = t[i];
    if (tid < 2) sb1[tid] = b1[tid];
    __syncthreads();

    const int lane = tid & 31;
    const int wave = tid >> 5;
    const int b    = blockIdx.x * (blockDim.x >> 5) + wave;
    if (b >= B_SZ) return;

    const int h0 = lane;
    const int h1 = lane + 32;
    const float* cb = coeffs + (size_t)b * NI * (4 * C_SZ);

    // ---- z0 = X0 @ W_init + b_init, X0 = coeffs[b, 0, 0:32] (the 'a' block)
    float z0 = b_init[h0];
    float z1 = b_init[h1];
#pragma unroll 8
    for (int c = 0; c < C_SZ; ++c) {
        const float xc = cb[c];                     // uniform across lanes
        z0 = fmaf(xc, sWinit[c * H_SZ + h0], z0);   // lanes -> consecutive banks
        z1 = fmaf(xc, sWinit[c * H_SZ + h1], z1);
    }
    z_hist[((size_t)b * L_SZ) * H_SZ + h0] = z0;
    z_hist[((size_t)b * L_SZ) * H_SZ + h1] = z1;

    // Per-lane W1 slices for the wave reduction
    const float w1a0 = sW1[h0 * 2 + 0], w1a1 = sW1[h0 * 2 + 1];
    const float w1b0 = sW1[h1 * 2 + 0], w1b1 = sW1[h1 * 2 + 1];
    const float bb10 = sb1[0], bb11 = sb1[1];

    for (int i = 0; i < NI; ++i) {
        const float* row = cb + (size_t)i * (4 * C_SZ);
        const float bi = row[C_SZ     + lane];    // b   (channel = lane)
        const float ci = row[2 * C_SZ + lane];    // 2c
        const float di = row[3 * C_SZ + lane];    // 3d
        const float dti   = sT[i + 1] - sT[i];
        const float hstep = dti * 0.5f;           // N_SUB = 2

        // g(z, s): vector field evaluation
        auto gfun = [&](float za, float zb, float s, float& oa, float& ob) {
            // h1 = relu(z @ W1 + b1): reduce over all 64 h across the wave
            float p0 = za * w1a0 + zb * w1b0;
            float p1 = za * w1a1 + zb * w1b1;
#pragma unroll
            for (int m = 1; m < 32; m <<= 1) {
                p0 += __shfl_xor(p0, m, 32);
                p1 += __shfl_xor(p1, m, 32);
            }
            const float u0 = fmaxf(p0 + bb10, 0.0f);   // wave-uniform
            const float u1 = fmaxf(p1 + bb11, 0.0f);   // wave-uniform
            // dXdt for channel c = lane
            const float dX = fmaf(fmaf(di, s, ci), s, bi);
            float a0 = 0.0f, a1 = 0.0f;
#pragma unroll
            for (int c = 0; c < C_SZ; ++c) {
                const float w  = __shfl(dX, c, 32);    // v_readlane broadcast
                const int  i0  = c * H_SZ + h0;        // bank = lane: conflict-free
                const int  i1  = c * H_SZ + h1;
                const float2 wv0 = sPack[i0];          // ds_load_b64, all 64 banks
                const float2 wv1 = sPack[i1];
                const float t0 = fast_tanh(fmaf(u0, wv0.x, fmaf(u1, wv0.y, sB2t[i0])));
                const float t1 = fast_tanh(fmaf(u0, wv1.x, fmaf(u1, wv1.y, sB2t[i1])));
                a0 = fmaf(t0, w, a0);
                a1 = fmaf(t1, w, a1);
            }
            oa = a0; ob = a1;
        };

        for (int j = 0; j < 2; ++j) {
            const float s0 = (float)j * hstep;
            float k1a, k1b, k2a, k2b, k3a, k3b, k4a, k4b;
            gfun(z0,                      z1,                      s0,                k1a, k1b);
            gfun(z0 + 0.5f * hstep * k1a, z1 + 0.5f * hstep * k1b, s0 + 0.5f * hstep, k2a, k2b);
            gfun(z0 + 0.5f * hstep * k2a, z1 + 0.5f * hstep * k2b, s0 + 0.5f * hstep, k3a, k3b);
            gfun(z0 + hstep * k3a,        z1 + hstep * k3b,        s0 + hstep,        k4a, k4b);
            const float w6 = hstep * (1.0f / 6.0f);
            z0 += w6 * (k1a + 2.0f * k2a + 2.0f * k3a + k4a);
            z1 += w6 * (k1b + 2.0f * k2b + 2.0f * k3b + k4b);
        }
        z_hist[((size_t)b * L_SZ + i + 1) * H_SZ + h0] = z0;
        z_hist[((size_t)b * L_SZ + i + 1) * H_SZ + h1] = z1;
    }
}

// ---------------------------------------------------------------------------
// Kernel 2: pred_y = tanh(z_T) @ W_read + b_read via V_WMMA_F32_16X16X4_F32.
// One wave per 16-row tile; K=64 in 16 chunks of 4.
// A (16x4 f32): lanes 0-15 -> M=lane, K={k0,k0+1}; lanes 16-31 -> K={k0+2,k0+3}
// B (4x16 f32): v0 = K=k0 (lanes0-15) / K=k0+2 (lanes16-31), v1 = K=k0+1/k0+3
// C/D (16x16):  VGPR r -> M=r (lanes0-15) / M=r+8 (lanes16-31), N=lane&15
// ---------------------------------------------------------------------------
__global__ __launch_bounds__(256) void ncde_readout_kernel(
    const float* __restrict__ z_hist,   // (B*L, 64)
    const float* __restrict__ W_read,   // (64, 16)
    const float* __restrict__ b_read,   // (16,)
    float* __restrict__ pred_y)         // (B*L, 16)
{
    __shared__ float sW[H_SZ * O_SZ];   // 1024 floats
    const int tid = threadIdx.x;
    for (int i = tid; i < H_SZ * O_SZ; i += blockDim.x) sW[i] = W_read[i];
    __syncthreads();

    const int lane  = tid & 31;
    const int wave  = tid >> 5;
    const int tile  = blockIdx.x * (blockDim.x >> 5) + wave;   // 16384 tiles
    const int m     = lane & 15;
    const int khalf = (lane >> 4) * 2;
    const size_t rowbase = ((size_t)tile * 16 + m) * H_SZ;
    const int col   = m;

    v8f acc = {0.f, 0.f, 0.f, 0.f, 0.f, 0.f, 0.f, 0.f};
#pragma unroll
    for (int ck = 0; ck < 16; ++ck) {
        const int kk = ck * 4 + khalf;
        v2f a = *reinterpret_cast<const v2f*>(z_hist + rowbase + kk);
        a.x = fast_tanh(a.x);
        a.y = fast_tanh(a.y);
        v2f bf;
        bf.x = sW[kk * O_SZ + col];
        bf.y = sW[(kk + 1) * O_SZ + col];
        acc = __builtin_amdgcn_wmma_f32_16x16x4_f32(
            /*neg_a=*/false, a, /*neg_b=*/false, bf,
            /*c_mod=*/(short)0, acc, /*reuse_a=*/false, /*reuse_b=*/false);
    }

    const float bcol = b_read[col];
#pragma unroll
    for (int r = 0; r < 8; ++r) {
        const int mr = (lane < 16) ? r : (r + 8);
        pred_y[((size_t)tile * 16 + mr) * O_SZ + col] = acc[r] + bcol;
    }
}

// ---------------------------------------------------------------------------
// Kernel 3: new_coeffs = hermite_cubic_coeffs(pred_y, t), packed [a,b,2c,3d].
// One thread per (b, interval, o).
// ---------------------------------------------------------------------------
__global__ __launch_bounds__(256) void ncde_hermite_kernel(
    const float* __restrict__ y,        // pred_y (B, L, O)
    const float* __restrict__ t,        // (L,)
    float* __restrict__ out)            // (B, 127, 64)
{
    const int tid   = blockIdx.x * blockDim.x + threadIdx.x;
    const int total = B_SZ * NI * O_SZ;
    if (tid >= total) return;
    const int o   = tid & (O_SZ - 1);
    const int rem = tid >> 4;
    const int i   = rem % NI;
    const int b   = rem / NI;

    const size_t base = ((size_t)b * L_SZ + i) * O_SZ + o;
    const float x0 = y[base];
    const float x1 = y[base + O_SZ];
    const float dt = t[i + 1] - t[i];
    const float delta = x1 - x0;
    const float sec   = delta / dt;
    float d0;
    if (i == 0) {
        d0 = sec;                                  // derivs[0] = secants[0]
    } else {
        const float xm1 = y[base - O_SZ];
        const float dtp = t[i] - t[i - 1];
        d0 = (x0 - xm1) / dtp;                     // derivs[i] = secants[i-1]
    }
    const float d1 = sec;
    const float two_c   = 2.0f * (3.0f * delta / dt - 2.0f * d0 - d1) / dt;
    const float three_d = 3.0f * (-2.0f * delta / dt + d0 + d1) / (dt * dt);

    const size_t ob = ((size_t)b * NI + i) * (4 * O_SZ);
    out[ob + o]              = x0;       // a
    out[ob + O_SZ + o]       = d0;       // b
    out[ob + 2 * O_SZ + o]   = two_c;    // 2c
    out[ob + 3 * O_SZ + o]   = three_d;  // 3d
}

// ---------------------------------------------------------------------------
extern "C" void kernel_launch(void* const* d_in, const int* in_sizes, int n_in,
                              void* d_out, int out_size, void* d_ws, size_t ws_size,
                              hipStream_t stream) {
    const float* coeffs = (const float*)d_in[0];
    const float* t      = (const float*)d_in[1];
    const float* W_init = (const float*)d_in[2];
    const float* b_init = (const float*)d_in[3];
    const float* W1     = (const float*)d_in[4];
    const float* b1     = (const float*)d_in[5];
    const float* W2     = (const float*)d_in[6];
    const float* b2     = (const float*)d_in[7];
    const float* W_read = (const float*)d_in[8];
    const float* b_read = (const float*)d_in[9];

    float* pred_y     = (float*)d_out;                               // B*L*O
    float* new_coeffs = pred_y + (size_t)B_SZ * L_SZ * O_SZ;         // B*127*64
    float* z_hist     = (float*)d_ws;                                // B*L*H f32 (64 MiB)

    // Scan: 1 wave per batch element; 8 waves (256 threads) per block.
    ncde_scan_kernel<<<B_SZ / 8, 256, 0, stream>>>(
        coeffs, t, W_init, b_init, W1, b1, W2, b2, z_hist);

    // Readout GEMM via f32 WMMA: (B*L x 64) @ (64 x 16); 1 wave per 16-row tile.
    const int tiles = (B_SZ * L_SZ) / 16;   // 16384
    ncde_readout_kernel<<<tiles / 8, 256, 0, stream>>>(
        z_hist, W_read, b_read, pred_y);

    // Hermite coefficients of pred_y.
    const int total = B_SZ * NI * O_SZ;
    ncde_hermite_kernel<<<(total + 255) / 256, 256, 0, stream>>>(
        pred_y, t, new_coeffs);
}